// HBondWholePoseScoringModule_49761491091731
// MI455X (gfx1250) — compile-verified
//
#include <hip/hip_runtime.h>
#include <hip/hip_bf16.h>

typedef __attribute__((ext_vector_type(2))) float v2f;
typedef __attribute__((ext_vector_type(8))) float v8f;

// Problem constants (from the reference)
constexpr int P_   = 8;
constexpr int N_   = 160;
constexpr int A_   = 32;
constexpr int DS   = N_ * 4;   // 640 donor slots per pose
constexpr int AS   = N_ * 8;   // 1280 acceptor slots per pose
constexpr int DTIL = DS / 16;  // 40 donor tiles
constexpr int ATIL = AS / 16;  // 80 acceptor tiles
constexpr int NPOLY = 11;
constexpr int NPAIR = 256;
constexpr int PSTR  = 12;      // padded poly row stride (48B, 16B-aligned)

// ---------------------------------------------------------------------------
// Phase 1: densify donor/acceptor slot data per pose. Also zero the output.
// ---------------------------------------------------------------------------
__global__ void hb_gather(const float* __restrict__ coords,
                          const int*   __restrict__ blk_off,
                          const int*   __restrict__ blk_type,
                          const int*   __restrict__ n_donH,
                          const int*   __restrict__ n_acc,
                          const int*   __restrict__ donH_inds,
                          const int*   __restrict__ acc_inds,
                          const int*   __restrict__ donor_type,
                          const int*   __restrict__ acceptor_type,
                          const int*   __restrict__ acc_hyb,
                          float4* __restrict__ dpos, float4* __restrict__ apos,
                          float*  __restrict__ dhh,  float*  __restrict__ ahh,
                          int*    __restrict__ dmeta, int*   __restrict__ ameta,
                          float*  __restrict__ out)
{
    int idx = blockIdx.x * blockDim.x + threadIdx.x;
    if (idx < P_) out[idx] = 0.0f;

    const int nd = P_ * DS;
    if (idx < nd) {
        // donor slot
        int p = idx / DS, s = idx % DS;
        int b = s >> 2, j = s & 3;
        int bt  = blk_type[p * N_ + b];
        int off = blk_off [p * N_ + b];
        int ai  = donH_inds[bt * 4 + j];
        const float* cp = coords + ((size_t)p * (N_ * A_) + off + ai) * 3;
        float x = cp[0], y = cp[1], z = cp[2];
        dpos[idx] = make_float4(x, y, z, 0.0f);      // w=0 pads K=3 -> K=4
        dhh [idx] = x * x + y * y + z * z;
        int valid = (j < n_donH[bt]) ? 1 : 0;
        dmeta[idx] = (donor_type[bt * 4 + j] & 255) | (valid << 8);
    } else if (idx < nd + P_ * AS) {
        // acceptor slot
        int r = idx - nd;
        int p = r / AS, t = r % AS;
        int b = t >> 3, k = t & 7;
        int bt  = blk_type[p * N_ + b];
        int off = blk_off [p * N_ + b];
        int ai  = acc_inds[bt * 8 + k];
        const float* cp = coords + ((size_t)p * (N_ * A_) + off + ai) * 3;
        float x = cp[0], y = cp[1], z = cp[2];
        apos[r] = make_float4(x, y, z, 0.0f);
        ahh [r] = x * x + y * y + z * z;
        int valid = (k < n_acc[bt]) ? 1 : 0;
        ameta[r] = (acceptor_type[bt * 8 + k] & 255) |
                   ((acc_hyb[bt * 8 + k] & 3) << 8) | (valid << 16);
    }
}

// ---------------------------------------------------------------------------
// Phase 2: per pose, 640x1280 cross inner products via V_WMMA_F32_16X16X4_F32,
// then polynomial + mask + weight + reduction. One wave per 16x16 tile.
// Grid: P_*DTIL blocks x 256 threads (8 waves); wave w handles acceptor tiles
// ta = w, w+8, ... (uniform per wave -> EXEC all ones for WMMA).
// Hot tables (poly coeffs, pair weights, minsep rows) staged in LDS.
// ---------------------------------------------------------------------------
__global__ __launch_bounds__(256) void hb_main(
        const float4* __restrict__ dpos, const float4* __restrict__ apos,
        const float*  __restrict__ dhh,  const float*  __restrict__ ahh,
        const int*    __restrict__ dmeta, const int*   __restrict__ ameta,
        const int*    __restrict__ minsep,
        const float*  __restrict__ pair_params,   // [256][3]
        const float*  __restrict__ polys,         // [256][11]
        const float*  __restrict__ gp,            // [4]
        float* __restrict__ out)
{
    __shared__ __align__(16) float s_poly[NPAIR * PSTR];   // 12 KB
    __shared__ float s_pp[NPAIR * 3];                      // 3 KB
    __shared__ int   s_ms[4 * N_];                         // 2.5 KB

    const int p    = blockIdx.x / DTIL;
    const int td   = blockIdx.x % DTIL;
    const int wave = threadIdx.x >> 5;
    const int lane = threadIdx.x & 31;
    const int hl   = lane >> 4;        // half-wave select (K-split / M-split)
    const int l15  = lane & 15;

    // --- cooperative LDS staging -------------------------------------------
    for (int i = threadIdx.x; i < NPAIR * NPOLY; i += 256) {
        int r = i / NPOLY, c = i % NPOLY;
        s_poly[r * PSTR + c] = polys[i];
    }
    for (int i = threadIdx.x; i < NPAIR * 3; i += 256)
        s_pp[i] = pair_params[i];
    // minsep rows for this tile's 4 donor blocks (b1 = td*4 .. td*4+3)
    for (int i = threadIdx.x; i < 4 * N_; i += 256) {
        int q = i / N_, b2 = i % N_;
        s_ms[i] = minsep[(p * N_ + td * 4 + q) * N_ + b2];
    }
    __syncthreads();

    const float gp0 = gp[0];
    const float gp1 = gp[1];

    // --- A operand: donor tile (16x4 f32, K padded with 0) ---
    const int sbase = td * 16;
    float4 dp = dpos[p * DS + sbase + l15];
    v2f aop;
    aop.x = hl ? dp.z : dp.x;
    aop.y = hl ? 0.0f : dp.y;

    // --- donor-side metadata hoisted out of the acceptor loop ---
    // C/D layout: lane element v corresponds to row m = v + 8*hl, col n = l15.
    float hhv[8];
    int   ptb[8];
    int   b1v[8];     // global b1 (for b1 != b2 test)
    int   b1l[8];     // local b1 in [0,4) for s_ms indexing
    int   dvl[8];
#pragma unroll
    for (int v = 0; v < 8; ++v) {
        int m  = v + 8 * hl;
        int s  = sbase + m;
        hhv[v] = dhh[p * DS + s];
        int dm = dmeta[p * DS + s];
        ptb[v] = (dm & 255) * 16;       // donor_type * NAT
        dvl[v] = (dm >> 8) & 1;
        b1v[v] = s >> 2;
        b1l[v] = m >> 2;
    }

    float local = 0.0f;

    for (int ta = wave; ta < ATIL; ta += 8) {
        const int tbase = ta * 16;
        const int t = tbase + l15;

        // --- B operand: acceptor tile (4x16 f32) ---
        float4 ap = apos[p * AS + t];
        v2f bop;
        bop.x = hl ? ap.z : ap.x;
        bop.y = hl ? 0.0f : ap.y;

        v8f c = {};
        // D = A x B (+0): 16x16 tile of H.Ac inner products
        c = __builtin_amdgcn_wmma_f32_16x16x4_f32(
                /*neg_a=*/false, aop, /*neg_b=*/false, bop,
                /*c_mod=*/(short)0, c, /*reuse_a=*/false, /*reuse_b=*/false);

        // acceptor-side per-column data
        const float aa  = ahh[p * AS + t];
        const int   am  = ameta[p * AS + t];
        const int   att = am & 255;
        const int   hyb = (am >> 8) & 3;
        const int   avl = (am >> 16) & 1;
        const int   b2  = t >> 3;

#pragma unroll
        for (int v = 0; v < 8; ++v) {
            // d^2 = |H|^2 + |Ac|^2 - 2 H.Ac  (clamped: rounding can go <0)
            float d2 = fmaxf(hhv[v] + aa - 2.0f * c[v], 0.0f) + 1e-12f;
            float d  = sqrtf(d2);

            int pt = ptb[v] + att;
            // 11 coeffs from LDS via three 16B reads (row padded to 48B)
            const float4* pc4 = (const float4*)(s_poly + pt * PSTR);
            float4 c0 = pc4[0], c1 = pc4[1], c2 = pc4[2];
            float e = c0.x;
            e = fmaf(e, d, c0.y); e = fmaf(e, d, c0.z); e = fmaf(e, d, c0.w);
            e = fmaf(e, d, c1.x); e = fmaf(e, d, c1.y); e = fmaf(e, d, c1.z);
            e = fmaf(e, d, c1.w); e = fmaf(e, d, c2.x); e = fmaf(e, d, c2.y);
            e = fmaf(e, d, c2.z);
            e = fmaxf(fminf(e, 0.0f), gp1);

            float w  = s_pp[pt * 3 + hyb];
            int   ms = s_ms[b1l[v] * N_ + b2];
            bool  ok = (dvl[v] != 0) & (avl != 0) & (b1v[v] != b2) &
                       (ms >= 4) & (d < gp0);
            local += ok ? w * e : 0.0f;
        }
    }

    // wave32 reduction, then one atomic per wave
#pragma unroll
    for (int off = 16; off > 0; off >>= 1)
        local += __shfl_xor(local, off, 32);
    if (lane == 0) atomicAdd(&out[p], local);
}

// ---------------------------------------------------------------------------
extern "C" void kernel_launch(void* const* d_in, const int* in_sizes, int n_in,
                              void* d_out, int out_size, void* d_ws, size_t ws_size,
                              hipStream_t stream)
{
    (void)in_sizes; (void)n_in; (void)out_size; (void)ws_size;

    const float* coords        = (const float*)d_in[0];
    // d_in[1] block_pair_dispatch_indices: full P x N x N meshgrid -> regenerated implicitly
    const int*   blk_off       = (const int*)d_in[2];
    const int*   blk_type      = (const int*)d_in[3];
    const int*   minsep        = (const int*)d_in[4];
    const int*   n_donH        = (const int*)d_in[5];
    const int*   n_acc         = (const int*)d_in[6];
    const int*   donH_inds     = (const int*)d_in[7];
    const int*   acc_inds      = (const int*)d_in[8];
    const int*   donor_type    = (const int*)d_in[9];
    const int*   acceptor_type = (const int*)d_in[10];
    const int*   acc_hyb       = (const int*)d_in[11];
    const float* pair_params   = (const float*)d_in[12];
    const float* polys         = (const float*)d_in[13];
    const float* gp            = (const float*)d_in[14];
    float* out = (float*)d_out;

    // Workspace layout (16B-aligned chunks)
    char* ws = (char*)d_ws;
    size_t o = 0;
    float4* dpos  = (float4*)(ws + o); o += (size_t)P_ * DS * sizeof(float4);
    float4* apos  = (float4*)(ws + o); o += (size_t)P_ * AS * sizeof(float4);
    float*  dhh   = (float*) (ws + o); o += (size_t)P_ * DS * sizeof(float);
    float*  ahh   = (float*) (ws + o); o += (size_t)P_ * AS * sizeof(float);
    int*    dmeta = (int*)   (ws + o); o += (size_t)P_ * DS * sizeof(int);
    int*    ameta = (int*)   (ws + o); o += (size_t)P_ * AS * sizeof(int);

    const int total = P_ * (DS + AS);
    hb_gather<<<(total + 255) / 256, 256, 0, stream>>>(
        coords, blk_off, blk_type, n_donH, n_acc, donH_inds, acc_inds,
        donor_type, acceptor_type, acc_hyb,
        dpos, apos, dhh, ahh, dmeta, ameta, out);

    hb_main<<<P_ * DTIL, 256, 0, stream>>>(
        dpos, apos, dhh, ahh, dmeta, ameta,
        minsep, pair_params, polys, gp, out);
}